// Attn_43439299231871
// MI455X (gfx1250) — compile-verified
//
#include <hip/hip_runtime.h>
#include <hip/hip_bf16.h>
#include <math.h>

#define HIDDEN   1024
#define SEQ_LEN  32768

typedef __attribute__((ext_vector_type(2))) float v2f;
typedef __attribute__((ext_vector_type(8))) float v8f;

// ---------------------------------------------------------------------------
// Kernel 1: energies[s] = sum_h enc[s,h] * hidden[h] via V_WMMA_F32_16X16X4_F32
//
// One wave per 16-row tile. A fragment (16x4 fp32, 2 VGPRs):
//   lane l holds row M = l&15; VGPR0 = A[M, k + 2*(l>>4)], VGPR1 = A[M, k+2*(l>>4)+1]
// B fragment: B[kk, n] = hidden[k+kk] broadcast over all 16 columns n, so every
// column of D accumulates the same 16 energies (no column masking needed).
// D (16x16 fp32, 8 VGPRs): VGPR i -> M = i + 8*(lane>>4), N = lane&15.
// ---------------------------------------------------------------------------
__global__ __launch_bounds__(256)
void attn_energies_wmma(const float* __restrict__ enc,
                        const float* __restrict__ hidden,
                        float* __restrict__ energies) {
    const int lane  = threadIdx.x & 31;
    const int wave  = threadIdx.x >> 5;               // 8 waves / block (wave32)
    const int tile  = blockIdx.x * 8 + wave;          // 16-row tile
    const int row0  = tile * 16;
    const int m     = lane & 15;                      // row this lane carries in A
    const int khalf = (lane >> 4) * 2;                // lanes 16-31 carry K+2,K+3

    const float* __restrict__ arow = enc + (size_t)(row0 + m) * HIDDEN + khalf;
    const float* __restrict__ hk   = hidden + khalf;

    v8f acc = {};
    #pragma unroll 4
    for (int k = 0; k < HIDDEN; k += 4) {
        v2f a, b;
        a.x = arow[k];              // A[m, k+khalf]      (merges to global_load_b64)
        a.y = arow[k + 1];          // A[m, k+khalf+1]
        b.x = hk[k];                // hidden[k+khalf]  broadcast over N
        b.y = hk[k + 1];            // hidden[k+khalf+1]
        acc = __builtin_amdgcn_wmma_f32_16x16x4_f32(
            /*neg_a=*/false, a, /*neg_b=*/false, b,
            /*c_mod=*/(short)0, acc, /*reuse_a=*/false, /*reuse_b=*/false);
    }

    // Every column of D holds the energies; lane 0 has rows row0+0..7 in acc[0..7],
    // lane 16 has rows row0+8..15 in acc[0..7].
    if ((lane & 15) == 0) {
        const int base = row0 + 8 * (lane >> 4);
        #pragma unroll
        for (int i = 0; i < 8; ++i)
            energies[base + i] = acc[i];
    }
}

// ---------------------------------------------------------------------------
// Kernel 2: softmax over 32768 energies, single 1024-thread block.
// Three grid-stride passes (max, sum-exp, normalize) + LDS tree reductions.
// 3 x 128KB of traffic -- negligible vs the 128MB matvec read.
// ---------------------------------------------------------------------------
__global__ __launch_bounds__(1024)
void attn_softmax(const float* __restrict__ e, float* __restrict__ out) {
    __shared__ float red[1024];
    const int tid = threadIdx.x;

    // Pass 1: global max
    float mx = -INFINITY;
    for (int i = tid; i < SEQ_LEN; i += 1024) mx = fmaxf(mx, e[i]);
    red[tid] = mx;
    __syncthreads();
    for (int s = 512; s > 0; s >>= 1) {
        if (tid < s) red[tid] = fmaxf(red[tid], red[tid + s]);
        __syncthreads();
    }
    mx = red[0];
    __syncthreads();

    // Pass 2: sum of exp
    float sum = 0.0f;
    for (int i = tid; i < SEQ_LEN; i += 1024) sum += __expf(e[i] - mx);
    red[tid] = sum;
    __syncthreads();
    for (int s = 512; s > 0; s >>= 1) {
        if (tid < s) red[tid] += red[tid + s];
        __syncthreads();
    }
    const float inv = 1.0f / red[0];

    // Pass 3: write normalized probabilities
    for (int i = tid; i < SEQ_LEN; i += 1024)
        out[i] = __expf(e[i] - mx) * inv;
}

extern "C" void kernel_launch(void* const* d_in, const int* in_sizes, int n_in,
                              void* d_out, int out_size, void* d_ws, size_t ws_size,
                              hipStream_t stream) {
    const float* hidden = (const float*)d_in[0];   // [1024]
    const float* enc    = (const float*)d_in[1];   // [32768, 1024]
    float*       out    = (float*)d_out;           // [1, 1, 32768] flat
    float*       energ  = (float*)d_ws;            // 32768 floats of scratch

    // 32768 rows / 16 rows-per-wave / 8 waves-per-block = 256 blocks
    attn_energies_wmma<<<256, 256, 0, stream>>>(enc, hidden, energ);
    attn_softmax<<<1, 1024, 0, stream>>>(energ, out);
}